// HNPLinearAngleSum_33406255628979
// MI455X (gfx1250) — compile-verified
//
#include <hip/hip_runtime.h>

#define B 16
#define C16 16
#define H 512
#define NRC 176

typedef float v2f __attribute__((ext_vector_type(2)));
typedef float v8f __attribute__((ext_vector_type(8)));

// -------- input index map (setup_inputs dict order, params flattened leaf-wise)
enum {
  I_W0 = 0, I_W1, I_W2, I_B0, I_B1, I_B2, I_ANGLE,
  I_W0RPT_W, I_W0RPT_B, I_W0RB_W, I_W0RB_B,
  I_BIAS0_W, I_BIAS0_B, I_BIAS0RC_W, I_BIAS0RC_B,
  I_ANG0_W1, I_ANG0_B1, I_ANG0_W2, I_ANG0_B2,
  I_ANGB0_W1, I_ANGB0_B1, I_ANGB0_W2, I_ANGB0_B2,
  I_L1_W, I_L1_B, I_L1RC_W, I_L1RC_B, I_L1R_W, I_L1R_B, I_L1C_W, I_L1C_B,
  I_BIAS1_W, I_BIAS1_B, I_BIAS1RC_W, I_BIAS1RC_B,
  I_ANG1_W1, I_ANG1_B1, I_ANG1_W2, I_ANG1_B2,
  I_ANGB1_W1, I_ANGB1_B1, I_ANGB1_W2, I_ANGB1_B2,
  I_WFINCPT_W, I_WFINCPT_B, I_WFINCB_W, I_WFINCB_B,
  I_BFIN_W, I_BFIN_B,
  I_ANG2_W1, I_ANG2_B1, I_ANG2_W2, I_ANG2_B2,
  I_ANGB2_W1, I_ANGB2_B1, I_ANGB2_W2, I_ANGB2_B2,
  I_COUNT
};

struct Ptrs { const float* p[64]; };

// -------- workspace layout (float offsets)
#define WS_RM1   0          // row_m1 [B,16,512]  (w1 mean over H2)
#define WS_CM1   131072     // col_m1 [B,16,512]  (w1 mean over H1)
#define WS_CM0   262144     // col_m0 [B,16,512]  (w0 mean over N_IN)
#define WS_ADD   393216     // per-(b,o,h) additive term for zw1
#define WS_ROW   524288     // per-(b,o,w) additive term for zw1
#define WS_RCIN  655360     // rc_inp [B,176]
#define WS_HID1  658176     // ang1 hidden [B,128]
#define WS_S0    660224     // [B,48]
#define WS_SU0   660992     // [B,16]
#define WS_SZ1   661248
#define WS_SU1   661504
#define WS_SZ2   661760

// -------- output offsets (floats)
#define O_ZW0 0
#define O_ZW1 393216u
#define O_ZW2 4587520u
#define O_U0  4718592u
#define O_U1  4849664u
#define O_U2  4980736u

// ============ K1a: row_m1 (mean over H2) + rc_m1 ============
__global__ void k1a_rowm1(const float* __restrict__ w1, float* __restrict__ ws) {
  __shared__ float red[256];
  int bc = blockIdx.x, t = threadIdx.x;
  const float* base = w1 + (size_t)bc * H * H;
  float s1 = 0.f, s2 = 0.f;
  for (int hh = 0; hh < H; ++hh) {
    s1 += base[hh * H + t];
    s2 += base[hh * H + t + 256];
  }
  ws[WS_RM1 + bc * H + t]       = s1 * (1.f / 512.f);
  ws[WS_RM1 + bc * H + t + 256] = s2 * (1.f / 512.f);
  red[t] = s1 + s2;
  __syncthreads();
  for (int off = 128; off > 0; off >>= 1) {
    if (t < off) red[t] += red[t + off];
    __syncthreads();
  }
  if (t == 0)
    ws[WS_RCIN + (bc >> 4) * NRC + 16 + (bc & 15)] = red[0] * (1.f / (512.f * 512.f));
}

// ============ K1b: col_m1 (mean over H1), wave-level shuffles ============
__global__ void k1b_colm1(const float* __restrict__ w1, float* __restrict__ ws) {
  int bc = blockIdx.x;
  int lane = threadIdx.x & 31;
  int wv = threadIdx.x >> 5;
  const float* base = w1 + (size_t)bc * H * H;
  for (int h = wv * 64; h < wv * 64 + 64; ++h) {
    const float* row = base + h * H;
    float s = 0.f;
#pragma unroll
    for (int j = 0; j < 16; ++j) s += row[lane + 32 * j];
    for (int off = 16; off > 0; off >>= 1) s += __shfl_xor(s, off, 32);
    if (lane == 0) ws[WS_CM1 + bc * H + h] = s * (1.f / 512.f);
  }
}

// ============ K2: w0 / w2 / b0 / b1 / b2 stats -> CM0 + rc_inp ============
__global__ void k2_stats(Ptrs P, float* __restrict__ ws) {
  __shared__ float s[6];
  int bc = blockIdx.x, t = threadIdx.x;
  int b = bc >> 4, c = bc & 15;
  if (t < 6) s[t] = 0.f;
  __syncthreads();
  const float* w0b = P.p[I_W0] + (size_t)bc * H * 3;
  float r0 = 0, r1 = 0, r2 = 0, sb0 = 0, sb1 = 0, sw2 = 0;
#pragma unroll
  for (int rep = 0; rep < 2; ++rep) {
    int hh = t + rep * 256;
    float v0 = w0b[hh * 3 + 0], v1 = w0b[hh * 3 + 1], v2 = w0b[hh * 3 + 2];
    ws[WS_CM0 + bc * H + hh] = (v0 + v1 + v2) * (1.f / 3.f);
    r0 += v0; r1 += v1; r2 += v2;
    sb0 += P.p[I_B0][bc * H + hh];
    sb1 += P.p[I_B1][bc * H + hh];
    sw2 += P.p[I_W2][bc * H + hh];
  }
  atomicAdd(&s[0], r0); atomicAdd(&s[1], r1); atomicAdd(&s[2], r2);
  atomicAdd(&s[3], sb0); atomicAdd(&s[4], sb1); atomicAdd(&s[5], sw2);
  __syncthreads();
  if (t == 0) {
    float* rc = ws + WS_RCIN + b * NRC;
    float tw0 = s[0] + s[1] + s[2];
    float b2v = P.p[I_B2][bc];
    rc[c]          = tw0 * (1.f / (512.f * 3.f));   // rc_m0
    rc[32 + c]     = s[5] * (1.f / 512.f);          // rc_m2
    rc[48 + c]     = s[3] * (1.f / 512.f);          // b_m0
    rc[64 + c]     = s[4] * (1.f / 512.f);          // b_m1
    rc[80 + c]     = b2v;                           // b_m2
    rc[96 + c * 3 + 0] = s[0] * (1.f / 512.f);      // row_m0
    rc[96 + c * 3 + 1] = s[1] * (1.f / 512.f);
    rc[96 + c * 3 + 2] = s[2] * (1.f / 512.f);
    rc[144 + c]    = s[5] * (1.f / 512.f);          // col_m2
    rc[160 + c]    = b2v;                           // b2 flat
  }
}

// ============ K4: per-batch small MLPs + rc-linears + u2 ============
__global__ void k4_small(Ptrs P, float* __restrict__ ws, float* __restrict__ out) {
  __shared__ float rcv[NRC];
  __shared__ float ang[3];
  __shared__ float h0[128], hb0[128], h1[128], hb1[128], h2[128], hb2[64];
  int b = blockIdx.x, t = threadIdx.x;
  if (t < NRC) rcv[t] = ws[WS_RCIN + b * NRC + t];
  if (t < 3) ang[t] = P.p[I_ANGLE][b * 3 + t];
  __syncthreads();
  float x0 = ang[0], x1 = ang[1], x2 = ang[2];
  if (t < 128) {
    float v;
    v = P.p[I_ANG0_W1][t*3]*x0 + P.p[I_ANG0_W1][t*3+1]*x1 + P.p[I_ANG0_W1][t*3+2]*x2 + P.p[I_ANG0_B1][t];
    h0[t] = fmaxf(v, 0.f);
    v = P.p[I_ANG1_W1][t*3]*x0 + P.p[I_ANG1_W1][t*3+1]*x1 + P.p[I_ANG1_W1][t*3+2]*x2 + P.p[I_ANG1_B1][t];
    h1[t] = fmaxf(v, 0.f);
    ws[WS_HID1 + b * 128 + t] = h1[t];
    v = P.p[I_ANG2_W1][t*3]*x0 + P.p[I_ANG2_W1][t*3+1]*x1 + P.p[I_ANG2_W1][t*3+2]*x2 + P.p[I_ANG2_B1][t];
    h2[t] = fmaxf(v, 0.f);
  } else {
    int u = t - 128;
    float v;
    v = P.p[I_ANGB0_W1][u*3]*x0 + P.p[I_ANGB0_W1][u*3+1]*x1 + P.p[I_ANGB0_W1][u*3+2]*x2 + P.p[I_ANGB0_B1][u];
    hb0[u] = fmaxf(v, 0.f);
    v = P.p[I_ANGB1_W1][u*3]*x0 + P.p[I_ANGB1_W1][u*3+1]*x1 + P.p[I_ANGB1_W1][u*3+2]*x2 + P.p[I_ANGB1_B1][u];
    hb1[u] = fmaxf(v, 0.f);
    if (u < 64) {
      v = P.p[I_ANGB2_W1][u*3]*x0 + P.p[I_ANGB2_W1][u*3+1]*x1 + P.p[I_ANGB2_W1][u*3+2]*x2 + P.p[I_ANGB2_B1][u];
      hb2[u] = fmaxf(v, 0.f);
    }
  }
  __syncthreads();
  if (t < 48) {
    float s = P.p[I_ANG0_B2][t];
    for (int k = 0; k < 128; ++k) s += P.p[I_ANG0_W2][t*128+k] * h0[k];
    float r = P.p[I_W0RB_B][t];
    for (int k = 0; k < NRC; ++k) r += P.p[I_W0RB_W][t*NRC+k] * rcv[k];
    ws[WS_S0 + b * 48 + t] = P.p[I_W0RPT_B][t] + r + s;
  } else if (t >= 64 && t < 80) {
    int c = t - 64;
    float s = P.p[I_ANGB0_B2][c];
    for (int k = 0; k < 128; ++k) s += P.p[I_ANGB0_W2][c*128+k] * hb0[k];
    float r = P.p[I_BIAS0RC_B][c];
    for (int k = 0; k < NRC; ++k) r += P.p[I_BIAS0RC_W][c*NRC+k] * rcv[k];
    ws[WS_SU0 + b * 16 + c] = P.p[I_BIAS0_B][c] + r + s;
  } else if (t >= 96 && t < 112) {
    int c = t - 96;
    float r = P.p[I_L1RC_B][c];
    for (int k = 0; k < NRC; ++k) r += P.p[I_L1RC_W][c*NRC+k] * rcv[k];
    ws[WS_SZ1 + b * 16 + c] = P.p[I_L1_B][c] + r;
  } else if (t >= 128 && t < 144) {
    int c = t - 128;
    float s = P.p[I_ANGB1_B2][c];
    for (int k = 0; k < 128; ++k) s += P.p[I_ANGB1_W2][c*128+k] * hb1[k];
    float r = P.p[I_BIAS1RC_B][c];
    for (int k = 0; k < NRC; ++k) r += P.p[I_BIAS1RC_W][c*NRC+k] * rcv[k];
    ws[WS_SU1 + b * 16 + c] = P.p[I_BIAS1_B][c] + r + s;
  } else if (t >= 160 && t < 176) {
    int c = t - 160;
    float s = P.p[I_ANG2_B2][c];
    for (int k = 0; k < 128; ++k) s += P.p[I_ANG2_W2][c*128+k] * h2[k];
    float r = P.p[I_WFINCB_B][c];
    for (int k = 0; k < NRC; ++k) r += P.p[I_WFINCB_W][c*NRC+k] * rcv[k];
    ws[WS_SZ2 + b * 16 + c] = P.p[I_WFINCPT_B][c] + r + s;
  } else if (t >= 192 && t < 208) {
    int c = t - 192;
    float s = P.p[I_ANGB2_B2][c];
    for (int k = 0; k < 64; ++k) s += P.p[I_ANGB2_W2][c*64+k] * hb2[k];
    float r = P.p[I_BFIN_B][c];
    for (int k = 0; k < NRC; ++k) r += P.p[I_BFIN_W][c*NRC+k] * rcv[k];
    out[O_U2 + b * 16 + c] = r + s;  // u2
  }
}

// ============ K5: ang1 big GEMM via WMMA f32 16x16x4 ============
// out[b, j] = sum_k W2[j,k]*hid[b,k] + b2[j]   (M=16 batches, N=8192, K=128)
__global__ void k5_ang1(const float* __restrict__ W2, const float* __restrict__ b2j,
                        float* ws) {
  int lane = threadIdx.x & 31;
  int wid = blockIdx.x * 8 + (threadIdx.x >> 5);   // 0..511
  int j0 = wid * 16;
  int n = lane & 15;
  int half = lane >> 4;
  const float* hid = ws + WS_HID1;
  v8f acc;
  float bias = b2j[j0 + n];
#pragma unroll
  for (int r = 0; r < 8; ++r) acc[r] = bias;
#pragma unroll 4
  for (int k0 = 0; k0 < 128; k0 += 4) {
    v2f a  = *(const v2f*)(hid + n * 128 + k0 + half * 2);            // A: M=batch
    v2f bf = *(const v2f*)(W2 + (size_t)(j0 + n) * 128 + k0 + half * 2);
    acc = __builtin_amdgcn_wmma_f32_16x16x4_f32(false, a, false, bf, (short)0, acc,
                                                false, false);
  }
  float* ADDp = ws + WS_ADD;
#pragma unroll
  for (int r = 0; r < 8; ++r) {
    int m = r + half * 8;                 // batch index
    ADDp[m * 8192 + j0 + n] = acc[r];
  }
}

// ============ K6: col-boundary convs -> addbuf += colterm+Sz1; u1; zw2 ============
__global__ void k6_col(Ptrs P, float* __restrict__ ws, float* __restrict__ out) {
  __shared__ float wlc[16 * 64], wb1[16 * 64], wfin[16 * 32];
  __shared__ float sz1[16], su1[16], sz2[16];
  int b = blockIdx.x >> 2;
  int h = ((blockIdx.x & 3) << 7) + threadIdx.x;
  int t = threadIdx.x;
  for (int i = t; i < 1024; i += 128) { wlc[i] = P.p[I_L1C_W][i]; wb1[i] = P.p[I_BIAS1_W][i]; }
  for (int i = t; i < 512; i += 128) wfin[i] = P.p[I_WFINCPT_W][i];
  if (t < 16) {
    sz1[t] = ws[WS_SZ1 + b * 16 + t];
    su1[t] = ws[WS_SU1 + b * 16 + t];
    sz2[t] = ws[WS_SZ2 + b * 16 + t];
  }
  __syncthreads();
  float colv[48];  // rows: [col_m1(16), b1(16), w2(16)]; rows 48..63 of col_bd repeat w2
#pragma unroll
  for (int k = 0; k < 16; ++k) colv[k]      = ws[WS_CM1 + (b * 16 + k) * H + h];
#pragma unroll
  for (int k = 0; k < 16; ++k) colv[16 + k] = P.p[I_B1][(b * 16 + k) * H + h];
#pragma unroll
  for (int k = 0; k < 16; ++k) colv[32 + k] = P.p[I_W2][(b * 16 + k) * H + h];
  for (int o = 0; o < 16; ++o) {
    float a1 = 0.f, au = 0.f, a2 = 0.f;
#pragma unroll
    for (int k = 0; k < 48; ++k) { a1 += wlc[o*64+k] * colv[k]; au += wb1[o*64+k] * colv[k]; }
#pragma unroll
    for (int k = 48; k < 64; ++k) { float v = colv[k - 16]; a1 += wlc[o*64+k] * v; au += wb1[o*64+k] * v; }
#pragma unroll
    for (int k = 0; k < 16; ++k) a2 += wfin[o*32+k] * colv[32 + k];        // cpt rows 0-15: w2_f
#pragma unroll
    for (int k = 16; k < 32; ++k) a2 += wfin[o*32+k] * colv[k - 16];       // cpt rows 16-31: col_m1
    size_t idx = (size_t)(b * 16 + o) * H + h;
    ws[WS_ADD + idx] += a1 + sz1[o];
    out[O_U1 + idx]  = au + su1[o];
    out[O_ZW2 + idx] = a2 + sz2[o];
  }
}

// ============ K7: row-boundary convs -> rowbuf; zw0; u0 ============
__global__ void k7_row(Ptrs P, float* __restrict__ ws, float* __restrict__ out) {
  __shared__ float wr[48 * 80];   // w0_rpt
  __shared__ float wu[16 * 80];   // bias0
  __shared__ float wl[16 * 96];   // l1_r
  __shared__ float s0s[48], su0s[16], lrb[16];
  int b = blockIdx.x >> 2;
  int w = ((blockIdx.x & 3) << 7) + threadIdx.x;
  int t = threadIdx.x;
  for (int i = t; i < 3840; i += 128) wr[i] = P.p[I_W0RPT_W][i];
  for (int i = t; i < 1280; i += 128) wu[i] = P.p[I_BIAS0_W][i];
  for (int i = t; i < 1536; i += 128) wl[i] = P.p[I_L1R_W][i];
  if (t < 48) s0s[t] = ws[WS_S0 + b * 48 + t];
  if (t < 16) { su0s[t] = ws[WS_SU0 + b * 16 + t]; lrb[t] = P.p[I_L1R_B][t]; }
  __syncthreads();
  float vals[96];  // [row_m1(16), col_m0(16), b0(16), w0_t(48)]
#pragma unroll
  for (int k = 0; k < 16; ++k) vals[k]      = ws[WS_RM1 + (b * 16 + k) * H + w];
#pragma unroll
  for (int k = 0; k < 16; ++k) vals[16 + k] = ws[WS_CM0 + (b * 16 + k) * H + w];
#pragma unroll
  for (int k = 0; k < 16; ++k) vals[32 + k] = P.p[I_B0][(b * 16 + k) * H + w];
#pragma unroll
  for (int c = 0; c < 16; ++c)
#pragma unroll
    for (int n = 0; n < 3; ++n)
      vals[48 + c * 3 + n] = P.p[I_W0][((size_t)(b * 16 + c) * H + w) * 3 + n];
  // rpt(k): k<48 -> vals[48+k] (w0_t); 48..63 -> vals[k-48] (row_m1); 64..79 -> vals[k-32] (b0)
  for (int oo = 0; oo < 48; ++oo) {
    float z = s0s[oo];
#pragma unroll
    for (int k = 0; k < 48; ++k) z += wr[oo*80+k] * vals[48 + k];
#pragma unroll
    for (int k = 48; k < 64; ++k) z += wr[oo*80+k] * vals[k - 48];
#pragma unroll
    for (int k = 64; k < 80; ++k) z += wr[oo*80+k] * vals[k - 32];
    int c = oo / 3, n = oo - c * 3;
    out[O_ZW0 + ((size_t)(b * 16 + c) * H + w) * 3 + n] = z;
  }
  for (int c = 0; c < 16; ++c) {
    float u = su0s[c];
#pragma unroll
    for (int k = 0; k < 48; ++k) u += wu[c*80+k] * vals[48 + k];
#pragma unroll
    for (int k = 48; k < 64; ++k) u += wu[c*80+k] * vals[k - 48];
#pragma unroll
    for (int k = 64; k < 80; ++k) u += wu[c*80+k] * vals[k - 32];
    out[O_U0 + (size_t)(b * 16 + c) * H + w] = u;
    float rt = lrb[c];
#pragma unroll
    for (int k = 0; k < 96; ++k) rt += wl[c*96+k] * vals[k];
    ws[WS_ROW + (size_t)(b * 16 + c) * H + w] = rt;
  }
}

// ============ K8: zw1 = l1W x w1 (+ addbuf + rowbuf) via WMMA f32 16x16x4 ============
__global__ void k8_zw1(const float* __restrict__ w1, const float* __restrict__ l1W,
                       const float* __restrict__ ws, float* __restrict__ out) {
  int lane = threadIdx.x & 31;
  int wid = blockIdx.x * 8 + (threadIdx.x >> 5);
  int b = wid >> 9;
  int rem = wid & 511;
  int wbase = (rem >> 4) * 16;   // 32 w-tiles of 16
  int hstart = (rem & 15) * 32;  // 16 h-chunks of 32
  int n = lane & 15;
  int half = lane >> 4;
  // A fragments: l1W [o=16, i=16] row-major; A lane m holds row M=m
  v2f a0 = *(const v2f*)(l1W + n * 16 + 0  + half * 2);
  v2f a1 = *(const v2f*)(l1W + n * 16 + 4  + half * 2);
  v2f a2 = *(const v2f*)(l1W + n * 16 + 8  + half * 2);
  v2f a3 = *(const v2f*)(l1W + n * 16 + 12 + half * 2);
  float rowv[8];
  const float* ROWp = ws + WS_ROW + (size_t)(b * 16) * H;
  const float* ADDp = ws + WS_ADD + (size_t)(b * 16) * H;
#pragma unroll
  for (int r = 0; r < 8; ++r) {
    int oo = r + half * 8;
    rowv[r] = ROWp[oo * H + wbase + n];
  }
  const float* w1b = w1 + (size_t)(b * 16) * H * H;
  float* outp = out + O_ZW1 + (size_t)(b * 16) * H * H;
  for (int h = hstart; h < hstart + 32; ++h) {
    v8f acc;
#pragma unroll
    for (int r = 0; r < 8; ++r) {
      int oo = r + half * 8;
      acc[r] = rowv[r] + ADDp[oo * H + h];
    }
    {
      int i = 0 + half * 2;
      v2f bf; bf.x = w1b[(size_t)i*H*H + h*H + wbase + n]; bf.y = w1b[(size_t)(i+1)*H*H + h*H + wbase + n];
      acc = __builtin_amdgcn_wmma_f32_16x16x4_f32(false, a0, false, bf, (short)0, acc, false, false);
    }
    {
      int i = 4 + half * 2;
      v2f bf; bf.x = w1b[(size_t)i*H*H + h*H + wbase + n]; bf.y = w1b[(size_t)(i+1)*H*H + h*H + wbase + n];
      acc = __builtin_amdgcn_wmma_f32_16x16x4_f32(false, a1, false, bf, (short)0, acc, false, false);
    }
    {
      int i = 8 + half * 2;
      v2f bf; bf.x = w1b[(size_t)i*H*H + h*H + wbase + n]; bf.y = w1b[(size_t)(i+1)*H*H + h*H + wbase + n];
      acc = __builtin_amdgcn_wmma_f32_16x16x4_f32(false, a2, false, bf, (short)0, acc, false, false);
    }
    {
      int i = 12 + half * 2;
      v2f bf; bf.x = w1b[(size_t)i*H*H + h*H + wbase + n]; bf.y = w1b[(size_t)(i+1)*H*H + h*H + wbase + n];
      acc = __builtin_amdgcn_wmma_f32_16x16x4_f32(false, a3, false, bf, (short)0, acc, false, false);
    }
#pragma unroll
    for (int r = 0; r < 8; ++r) {
      int oo = r + half * 8;
      outp[(size_t)oo * H * H + (size_t)h * H + wbase + n] = acc[r];
    }
  }
}

extern "C" void kernel_launch(void* const* d_in, const int* in_sizes, int n_in,
                              void* d_out, int out_size, void* d_ws, size_t ws_size,
                              hipStream_t stream) {
  (void)in_sizes; (void)out_size; (void)ws_size;
  Ptrs P;
  for (int i = 0; i < 64; ++i) P.p[i] = (i < n_in) ? (const float*)d_in[i] : nullptr;
  float* ws = (float*)d_ws;
  float* out = (float*)d_out;
  const float* w1 = P.p[I_W1];

  k1a_rowm1<<<256, 256, 0, stream>>>(w1, ws);
  k1b_colm1<<<256, 256, 0, stream>>>(w1, ws);
  k2_stats <<<256, 256, 0, stream>>>(P, ws);
  k4_small <<<16, 256, 0, stream>>>(P, ws, out);
  k5_ang1  <<<64, 256, 0, stream>>>(P.p[I_ANG1_W2], P.p[I_ANG1_B2], ws);
  k6_col   <<<64, 128, 0, stream>>>(P, ws, out);
  k7_row   <<<64, 128, 0, stream>>>(P, ws, out);
  k8_zw1   <<<1024, 256, 0, stream>>>(w1, P.p[I_L1_W], ws, out);
}